// ActorCritic_53403623358941
// MI455X (gfx1250) — compile-verified
//
#include <hip/hip_runtime.h>
#include <hip/hip_bf16.h>
#include <math.h>

// ---------------------------------------------------------------------------
// ActorCritic forward for MI455X (gfx1250, wave32).
// Bandwidth-bound: ~600MB fp32 weights streamed once -> ~26us at 23.3 TB/s.
// WMMA f32 16x16x4 for skinny GEMMs (M in [10,32]); VALU float4 GEMV for the
// M=1 head layers. All WMMA K-loops are branch-free (padding rows are allowed
// to carry garbage: D-row m depends only on A-row m) and unrolled so weight
// loads pipeline ahead of the WMMA chain.
// ---------------------------------------------------------------------------

#define D512 512
#define NH 8
#define HD 64
#define EPSV 1e-5f

// --- d_in index map (setup_inputs dict order, params flattened) -------------
enum {
  IN_HAND = 0, IN_DRAW, IN_DISC, IN_CHAR, IN_MON, IN_ENERGY,
  P_EMB_W, P_EMB_B, P_INP_W, P_INP_B, P_OUTP_W, P_OUTP_B,
  P_LN1G, P_LN1B, P_FF1W, P_FF1B, P_FF2W, P_FF2B, P_LN2G, P_LN2B,
  P_OW1, P_OB1, P_OW2, P_OB2, P_LNGG, P_LNGB, P_LNACG, P_LNACB,
  P_AC_W1, P_AC_B1, P_AC_W2, P_AC_B2, P_AC_W3, P_AC_B3,
  P_AM_W1, P_AM_B1, P_AM_W2, P_AM_B2, P_AM_W3, P_AM_B3,
  P_AE_W1, P_AE_B1, P_AE_W2, P_AE_B2, P_AE_W3, P_AE_B3,
  P_CR_W1, P_CR_B1, P_CR_W2, P_CR_B2, P_CR_W3, P_CR_B3,
  IN_HSIZE, IN_MASK, NUM_IN
};

struct Ptrs { const void* p[NUM_IN]; };
#define PF(i) ((const float*)P.p[(i)])

// --- workspace layout (float offsets) ---------------------------------------
#define WS_XHAND 0         /* 10*512   */
#define WS_XDRAW 5120      /* 1024     */
#define WS_XDISC 6144      /* 1024     */
#define WS_OTHER 7168      /* 1024     */
#define WS_XGLOB 8192      /* 4096     */
#define WS_ACIN  12288     /* 10*4608  */
#define WS_ACH1  58368     /* 16*4608  */
#define WS_ACH2  132096    /* 16*4608  */
#define WS_H1    205824    /* 3*4096   */
#define WS_H2    218112    /* 3*4096   */

typedef __attribute__((ext_vector_type(2))) float v2f;
typedef __attribute__((ext_vector_type(8))) float v8f;

// ---------------------------------------------------------------------------
// Branch-free WMMA K-loop: 16x16 fp32 tile, D = A(16xK) * Wtile^T + acc.
// A: row-major, lda stride; all 16 rows must be READABLE (padding rows may
//    hold garbage - it only propagates to padding rows of D, never to valid
//    rows, since D[m] depends only on A[m]).
// B: 16 consecutive rows of a row-major [N,K] weight matrix.
// VGPR striping per CDNA5 ISA 7.12.2: lanes 0-15 hold K=k,k+1;
// lanes 16-31 hold K=k+2,k+3 (for both A and B operands).
// ---------------------------------------------------------------------------
template <int K, bool PREF>
__device__ __forceinline__ v8f wmma_kloop(const float* __restrict__ A, int lda,
                                          const float* __restrict__ B, int ldb,
                                          int lane) {
  v8f acc = {0.f, 0.f, 0.f, 0.f, 0.f, 0.f, 0.f, 0.f};
  const int m = lane & 15;
  const int half = lane >> 4;
  const float* ar = A + m * lda + half * 2;
  const float* br = B + m * ldb + half * 2;
#pragma unroll 8
  for (int k = 0; k < K; k += 4) {
    v2f a, b;
    a.x = ar[k]; a.y = ar[k + 1];
    b.x = br[k]; b.y = br[k + 1];
    if (PREF && ((k & 63) == 0)) __builtin_prefetch(br + k + 1024, 0, 0);
    acc = __builtin_amdgcn_wmma_f32_16x16x4_f32(
        /*neg_a=*/false, a, /*neg_b=*/false, b,
        /*c_mod=*/(short)0, acc, /*reuse_a=*/false, /*reuse_b=*/false);
  }
  return acc;
}

// C/D layout: VGPR j -> (M = j + 8*half, N = n0 + lane&15)
__device__ __forceinline__ void wmma_store(v8f acc, float* __restrict__ out, int ldo,
                                           int mvalid, int m0, int n0,
                                           const float* __restrict__ bias,
                                           int lane, bool relu) {
  const int n = n0 + (lane & 15);
  const int half = lane >> 4;
  const float bv = bias ? bias[n] : 0.f;
#pragma unroll
  for (int j = 0; j < 8; ++j) {
    int m = j + half * 8;
    if (m < mvalid) {
      float v = acc[j] + bv;
      if (relu) v = fmaxf(v, 0.f);
      out[(size_t)(m0 + m) * ldo + n] = v;
    }
  }
}

// --- wave / block reductions -------------------------------------------------
__device__ __forceinline__ float wave_sum(float x) {
#pragma unroll
  for (int o = 16; o > 0; o >>= 1) x += __shfl_xor(x, o, 32);
  return x;
}
__device__ __forceinline__ float wave_max(float x) {
#pragma unroll
  for (int o = 16; o > 0; o >>= 1) x = fmaxf(x, __shfl_xor(x, o, 32));
  return x;
}
__device__ __forceinline__ void block_reduce2(float& a, float& b) {
  __shared__ float ra[16], rb[16];
  const int lane = threadIdx.x & 31, wid = threadIdx.x >> 5;
  const int nw = (blockDim.x + 31) >> 5;
  a = wave_sum(a); b = wave_sum(b);
  if (lane == 0) { ra[wid] = a; rb[wid] = b; }
  __syncthreads();
  if (wid == 0) {
    float x = (lane < nw) ? ra[lane] : 0.f;
    float y = (lane < nw) ? rb[lane] : 0.f;
    x = wave_sum(x); y = wave_sum(y);
    if (lane == 0) { ra[0] = x; rb[0] = y; }
  }
  __syncthreads();
  a = ra[0]; b = rb[0];
  __syncthreads();
}

// row-wise LayerNorm on an LDS buffer [S,512], one wave per row
__device__ __forceinline__ void ln_rows(float* buf, int S, const float* g, const float* bt,
                                        int lane, int wid, int nw) {
  for (int s = wid; s < S; s += nw) {
    float* r = buf + s * D512;
    float sum = 0.f, sq = 0.f;
    for (int i = lane; i < D512; i += 32) { float t = r[i]; sum += t; sq += t * t; }
    sum = wave_sum(sum); sq = wave_sum(sq);
    const float mean = sum * (1.f / D512);
    const float var = sq * (1.f / D512) - mean * mean;
    const float inv = rsqrtf(var + EPSV);
    for (int i = lane; i < D512; i += 32) r[i] = (r[i] - mean) * inv * g[i] + bt[i];
  }
}

// ---------------------------------------------------------------------------
// Kernel A: embed + transformer block for one branch per workgroup.
// blockIdx 0=hand(S=10,key mask=hand_size), 1=draw(S=30), 2=disc(S=15).
// Dynamic LDS: 5 buffers of S*512 floats + 1024-float tail pad so WMMA A-tile
// over-reads (rows S..Sp-1) stay inside the allocation (max 311296 B < 320KB).
// ---------------------------------------------------------------------------
__global__ void branch_block_kernel(Ptrs P, float* __restrict__ ws) {
  extern __shared__ float sm[];
  const int b = blockIdx.x;
  const float* src; int S; int klen;
  if (b == 0)      { src = PF(IN_HAND); S = 10; klen = *((const int*)P.p[IN_HSIZE]); }
  else if (b == 1) { src = PF(IN_DRAW); S = 30; klen = 30; }
  else             { src = PF(IN_DISC); S = 15; klen = 15; }

  float* x  = sm;                // residual stream
  float* q  = x  + S * D512;
  float* k  = q  + S * D512;
  float* v  = k  + S * D512;
  float* ao = v  + S * D512;     // attention output; also staging for src

  const int tid = threadIdx.x, lane = tid & 31, wid = tid >> 5;
  const int nw = blockDim.x >> 5;              // 16 waves
  const int mtiles = (S + 15) >> 4;

  // 0. stage src rows into LDS (so padded-row WMMA reads never go OOB in global)
  for (int i = tid; i < S * 128; i += blockDim.x) ao[i] = src[i];
  __syncthreads();

  // 1. embedding: x = src[S,128] @ emb_w[512,128]^T + emb_b
  {
    const float* W = PF(P_EMB_W); const float* bi = PF(P_EMB_B);
    for (int t = wid; t < mtiles * 32; t += nw) {
      int mt = t >> 5, nt = t & 31;
      v8f acc = wmma_kloop<128, false>(ao + mt * 16 * 128, 128,
                                       W + nt * 16 * 128, 128, lane);
      wmma_store(acc, x, D512, S - mt * 16, mt * 16, nt * 16, bi, lane, false);
    }
  }
  __syncthreads();

  // 2. qkv = x @ in_proj_w[1536,512]^T + b
  {
    const float* W = PF(P_INP_W); const float* bi = PF(P_INP_B);
    const int ntiles = 1536 / 16;
    for (int t = wid; t < mtiles * ntiles; t += nw) {
      int mt = t / ntiles, nt = t % ntiles;
      v8f acc = wmma_kloop<D512, false>(x + mt * 16 * D512, D512,
                                        W + nt * 16 * D512, D512, lane);
      int o = nt * 16 + (lane & 15);
      float* dst = (o < 512) ? q : (o < 1024 ? k : v);
      int col = o & 511;
      float bv = bi[o];
      int half = lane >> 4;
#pragma unroll
      for (int j = 0; j < 8; ++j) {
        int m = mt * 16 + j + half * 8;
        if (m < S) dst[m * D512 + col] = acc[j] + bv;
      }
    }
  }
  __syncthreads();

  // 3. attention: one wave per (query s, head h)
  {
    const float scale = 0.125f;  // 1/sqrt(64)
    for (int p = wid; p < S * NH; p += nw) {
      int s = p / NH, h = p % NH;
      const float4* qrow = (const float4*)(q + s * D512 + h * HD);
      float sc;
      if (lane < S && lane < klen) {
        const float4* krow = (const float4*)(k + lane * D512 + h * HD);
        float dot = 0.f;
#pragma unroll
        for (int d = 0; d < HD / 4; ++d) {
          float4 qv = qrow[d], kv = krow[d];
          dot += qv.x * kv.x + qv.y * kv.y + qv.z * kv.z + qv.w * kv.w;
        }
        sc = dot * scale;
      } else sc = -__builtin_inff();
      float mx = wave_max(sc);
      float e = __expf(sc - mx);
      if (!(lane < S && lane < klen)) e = 0.f;
      float se = wave_sum(e);
      float a = e / se;
      for (int d = lane; d < HD; d += 32) {
        float accv = 0.f;
        for (int t2 = 0; t2 < S; ++t2) {
          float w = __shfl(a, t2, 32);
          accv += w * v[t2 * D512 + h * HD + d];
        }
        ao[s * D512 + h * HD + d] = accv;
      }
    }
  }
  __syncthreads();

  // 4. out_proj + residual -> q, then LN1
  {
    const float* W = PF(P_OUTP_W); const float* bi = PF(P_OUTP_B);
    for (int t = wid; t < mtiles * 32; t += nw) {
      int mt = t >> 5, nt = t & 31;
      v8f acc = wmma_kloop<D512, false>(ao + mt * 16 * D512, D512,
                                        W + nt * 16 * D512, D512, lane);
      int n = nt * 16 + (lane & 15), half = lane >> 4;
#pragma unroll
      for (int j = 0; j < 8; ++j) {
        int m = mt * 16 + j + half * 8;
        if (m < S) q[m * D512 + n] = x[m * D512 + n] + acc[j] + bi[n];
      }
    }
  }
  __syncthreads();
  ln_rows(q, S, PF(P_LN1G), PF(P_LN1B), lane, wid, nw);
  __syncthreads();

  // 5. FF: hidden = relu(q @ ff1^T + b1) -> k ; out = q + hidden @ ff2^T + b2 -> v ; LN2
  {
    const float* W = PF(P_FF1W); const float* bi = PF(P_FF1B);
    for (int t = wid; t < mtiles * 32; t += nw) {
      int mt = t >> 5, nt = t & 31;
      v8f acc = wmma_kloop<D512, false>(q + mt * 16 * D512, D512,
                                        W + nt * 16 * D512, D512, lane);
      wmma_store(acc, k, D512, S - mt * 16, mt * 16, nt * 16, bi, lane, true);
    }
  }
  __syncthreads();
  {
    const float* W = PF(P_FF2W); const float* bi = PF(P_FF2B);
    for (int t = wid; t < mtiles * 32; t += nw) {
      int mt = t >> 5, nt = t & 31;
      v8f acc = wmma_kloop<D512, false>(k + mt * 16 * D512, D512,
                                        W + nt * 16 * D512, D512, lane);
      int n = nt * 16 + (lane & 15), half = lane >> 4;
#pragma unroll
      for (int j = 0; j < 8; ++j) {
        int m = mt * 16 + j + half * 8;
        if (m < S) v[m * D512 + n] = q[m * D512 + n] + acc[j] + bi[n];
      }
    }
  }
  __syncthreads();
  ln_rows(v, S, PF(P_LN2G), PF(P_LN2B), lane, wid, nw);
  __syncthreads();

  // 6. epilogue
  if (b == 0) {
    for (int i = tid; i < 10 * D512; i += blockDim.x) ws[WS_XHAND + i] = v[i];
  } else {
    float* outp = ws + (b == 1 ? WS_XDRAW : WS_XDISC);
    const float invS = 1.f / (float)S;
    for (int j = tid; j < D512; j += blockDim.x) {
      float s1 = 0.f;
      for (int s = 0; s < S; ++s) s1 += v[s * D512 + j];
      outp[j] = s1 * invS;
      outp[D512 + j] = s1;
    }
  }
}

// ---------------------------------------------------------------------------
// Kernel B: "other" MLP 120 -> 1024 -> 1024, ReLU after each linear.
// ---------------------------------------------------------------------------
__global__ void other_mlp_kernel(Ptrs P, float* __restrict__ ws) {
  __shared__ float in[120];
  __shared__ float h1[1024];
  const int tid = threadIdx.x;  // 1024 threads
  if (tid < 48)       in[tid] = PF(IN_CHAR)[tid];
  else if (tid < 112) in[tid] = PF(IN_MON)[tid - 48];
  else if (tid < 120) in[tid] = PF(IN_ENERGY)[tid - 112];
  __syncthreads();
  {
    const float* w = PF(P_OW1);
    float acc = PF(P_OB1)[tid];
    for (int kk = 0; kk < 120; ++kk) acc += in[kk] * w[tid * 120 + kk];
    h1[tid] = fmaxf(acc, 0.f);
  }
  __syncthreads();
  {
    const float* w = PF(P_OW2);
    float acc = PF(P_OB2)[tid];
    for (int kk = 0; kk < 1024; ++kk) acc += h1[kk] * w[tid * 1024 + kk];
    ws[WS_OTHER + tid] = fmaxf(acc, 0.f);
  }
}

// ---------------------------------------------------------------------------
// Kernel C: x_global assembly + LN(lng), then ac_in rows + LN(lnac).
// ---------------------------------------------------------------------------
__global__ void assemble_kernel(Ptrs P, float* __restrict__ ws) {
  __shared__ float g[4096];
  __shared__ float gs[2];  // sum, sumsq of normalized g
  const int tid = threadIdx.x;  // 512 threads

  for (int j = tid; j < 512; j += blockDim.x) {
    float s1 = 0.f;
    for (int s = 0; s < 10; ++s) s1 += ws[WS_XHAND + s * D512 + j];
    g[j] = s1 * 0.1f;
    g[512 + j] = s1;
  }
  for (int j = tid; j < 1024; j += blockDim.x) g[1024 + j] = ws[WS_XDRAW + j];
  for (int j = tid; j < 1024; j += blockDim.x) g[2048 + j] = ws[WS_XDISC + j];
  for (int j = tid; j < 1024; j += blockDim.x) g[3072 + j] = ws[WS_OTHER + j];
  __syncthreads();

  float sum = 0.f, sq = 0.f;
  for (int i = tid; i < 4096; i += blockDim.x) { float t = g[i]; sum += t; sq += t * t; }
  block_reduce2(sum, sq);
  {
    const float mean = sum * (1.f / 4096.f);
    const float var = sq * (1.f / 4096.f) - mean * mean;
    const float inv = rsqrtf(var + EPSV);
    const float* gg = PF(P_LNGG); const float* gb = PF(P_LNGB);
    for (int i = tid; i < 4096; i += blockDim.x) {
      float t = (g[i] - mean) * inv * gg[i] + gb[i];
      g[i] = t;
      ws[WS_XGLOB + i] = t;
    }
  }
  __syncthreads();

  // stats of normalized x_global (shared prefix of every ac_in row)
  float s1 = 0.f, s2 = 0.f;
  for (int i = tid; i < 4096; i += blockDim.x) { float t = g[i]; s1 += t; s2 += t * t; }
  block_reduce2(s1, s2);
  if (tid == 0) { gs[0] = s1; gs[1] = s2; }
  __syncthreads();

  const float* lag = PF(P_LNACG); const float* lab = PF(P_LNACB);
  for (int s = 0; s < 10; ++s) {
    const float* xr = ws + WS_XHAND + s * D512;
    float a1 = 0.f, a2 = 0.f;
    for (int i = tid; i < 512; i += blockDim.x) { float t = xr[i]; a1 += t; a2 += t * t; }
    block_reduce2(a1, a2);
    const float mean = (gs[0] + a1) * (1.f / 4608.f);
    const float var = (gs[1] + a2) * (1.f / 4608.f) - mean * mean;
    const float inv = rsqrtf(var + EPSV);
    float* orow = ws + WS_ACIN + s * 4608;
    for (int i = tid; i < 4096; i += blockDim.x)
      orow[i] = (g[i] - mean) * inv * lag[i] + lab[i];
    for (int i = tid; i < 512; i += blockDim.x)
      orow[4096 + i] = (xr[i] - mean) * inv * lag[4096 + i] + lab[4096 + i];
    __syncthreads();
  }
}

// ---------------------------------------------------------------------------
// Kernel D: ac head layer, out[16,4608] = relu(in[10..,4608] @ W[4608,4608]^T + b).
// WMMA f32 16x16x4, fully branch-free: LDS tile holds 16 rows (rows 10-15 are
// don't-care garbage that only feeds don't-care output rows). 4 waves/WG, one
// 16-wide N tile per wave, weight rows streamed contiguously with prefetch.
// ---------------------------------------------------------------------------
__global__ void ac_layer_kernel(const float* __restrict__ inp,
                                const float* __restrict__ W,
                                const float* __restrict__ bias,
                                float* __restrict__ outp) {
  extern __shared__ float sa[];  // 16*4608 floats = 294912 B
  const int tid = threadIdx.x, lane = tid & 31, wid = tid >> 5;
  for (int i = tid; i < 10 * 4608; i += blockDim.x) sa[i] = inp[i];
  __syncthreads();
  const int nt = blockIdx.x * 4 + wid;  // 288 tiles total
  v8f acc = wmma_kloop<4608, true>(sa, 4608, W + (size_t)nt * 16 * 4608, 4608, lane);
  wmma_store(acc, outp, 4608, 16, 0, nt * 16, bias, lane, true);
}

// ---------------------------------------------------------------------------
// Kernel E: streaming GEMV y[n] = (relu)(b[n] + x . W[n,:]); one wave per row.
// K must be a multiple of 128 (here 4096). Prefetch every 4th step.
// ---------------------------------------------------------------------------
__global__ void gemv_kernel(const float* __restrict__ x, const float* __restrict__ W,
                            const float* __restrict__ bias, float* __restrict__ y,
                            int N, int K, int relu) {
  const int lane = threadIdx.x & 31, wid = threadIdx.x >> 5;
  const int row = blockIdx.x * (blockDim.x >> 5) + wid;
  if (row >= N) return;
  const float4* wr = (const float4*)(W + (size_t)row * K);
  const float4* xv = (const float4*)x;
  const int nit = K >> 7;  // (K/4)/32 iterations
  float acc = 0.f;
#pragma unroll 4
  for (int it = 0; it < nit; ++it) {
    int i = lane + it * 32;
    float4 wv = wr[i];
    float4 xx = xv[i];
    if ((it & 3) == 0) __builtin_prefetch(wr + i + 128, 0, 0);
    acc += wv.x * xx.x + wv.y * xx.y + wv.z * xx.z + wv.w * xx.w;
  }
  acc = wave_sum(acc);
  if (lane == 0) {
    float r = acc + bias[row];
    if (relu) r = fmaxf(r, 0.f);
    y[row] = r;
  }
}

// ---------------------------------------------------------------------------
// Kernel F: final projections, logit assembly, mask + softmax, value.
// ---------------------------------------------------------------------------
__global__ void final_kernel(Ptrs P, float* __restrict__ ws, float* __restrict__ outp) {
  __shared__ float lg[23];
  const int lane = threadIdx.x & 31, wid = threadIdx.x >> 5;  // 256 thr, 8 waves
  for (int p = wid; p < 23; p += 8) {
    const float* a; const float* w; int K; float bv;
    if (p < 20) {
      int s = p % 10, i = p / 10;
      a = ws + WS_ACH2 + s * 4608; w = PF(P_AC_W3) + i * 4608; K = 4608; bv = PF(P_AC_B3)[i];
    } else if (p == 20) { a = ws + WS_H2;        w = PF(P_AM_W3); K = 4096; bv = PF(P_AM_B3)[0]; }
    else if (p == 21)   { a = ws + WS_H2 + 4096; w = PF(P_AE_W3); K = 4096; bv = PF(P_AE_B3)[0]; }
    else                { a = ws + WS_H2 + 8192; w = PF(P_CR_W3); K = 4096; bv = PF(P_CR_B3)[0]; }
    float acc = 0.f;
    for (int i2 = lane; i2 < K; i2 += 32) acc += a[i2] * w[i2];
    acc = wave_sum(acc);
    if (lane == 0) {
      int idx = p;
      if (p < 20) { int s = p % 10, i = p / 10; idx = i * 10 + s; }  // lp_card.T.reshape(-1)
      lg[idx] = acc + bv;
    }
  }
  __syncthreads();
  if (wid == 0) {
    const int* mask = (const int*)P.p[IN_MASK];
    float vlog = (lane < 22) ? ((mask[lane] == 0) ? -__builtin_inff() : lg[lane])
                             : -__builtin_inff();
    float mx = wave_max(vlog);
    float e = (lane < 22) ? __expf(vlog - mx) : 0.f;
    float se = wave_sum(e);
    if (lane < 22) outp[lane] = e / se;
    if (lane == 0) outp[22] = lg[22];  // value (cr head)
  }
}

// ---------------------------------------------------------------------------
extern "C" void kernel_launch(void* const* d_in, const int* in_sizes, int n_in,
                              void* d_out, int out_size, void* d_ws, size_t ws_size,
                              hipStream_t stream) {
  (void)in_sizes; (void)out_size; (void)ws_size;
  Ptrs P;
  for (int i = 0; i < NUM_IN && i < n_in; ++i) P.p[i] = d_in[i];
  float* ws = (float*)d_ws;
  float* out = (float*)d_out;

  // 1. three transformer branches (hand/draw/disc), one WG each
  //    5 * 30 * 512 floats + 1024-float tail pad for A-tile over-reads
  const size_t ldsA = ((size_t)5 * 30 * D512 + 1024) * sizeof(float);  // 311296 B
  branch_block_kernel<<<3, 512, ldsA, stream>>>(P, ws);

  // 2. other-features MLP
  other_mlp_kernel<<<1, 1024, 0, stream>>>(P, ws);

  // 3. x_global + ac_in assembly/LN
  assemble_kernel<<<1, 512, 0, stream>>>(P, ws);

  // 4. ac head big layers (WMMA, 288 N-tiles / 4 waves per WG)
  const size_t ldsD = (size_t)16 * 4608 * sizeof(float);  // 294912 B
  ac_layer_kernel<<<72, 128, ldsD, stream>>>(ws + WS_ACIN, PF(P_AC_W1), PF(P_AC_B1), ws + WS_ACH1);
  ac_layer_kernel<<<72, 128, ldsD, stream>>>(ws + WS_ACH1, PF(P_AC_W2), PF(P_AC_B2), ws + WS_ACH2);

  // 5. am/ae/cr head layers (streaming GEMV, 8 rows per 256-thread WG)
  gemv_kernel<<<512, 256, 0, stream>>>(ws + WS_XGLOB, PF(P_AM_W1), PF(P_AM_B1), ws + WS_H1 + 0,    4096, 4096, 1);
  gemv_kernel<<<512, 256, 0, stream>>>(ws + WS_XGLOB, PF(P_AE_W1), PF(P_AE_B1), ws + WS_H1 + 4096, 4096, 4096, 1);
  gemv_kernel<<<512, 256, 0, stream>>>(ws + WS_XGLOB, PF(P_CR_W1), PF(P_CR_B1), ws + WS_H1 + 8192, 4096, 4096, 1);
  gemv_kernel<<<512, 256, 0, stream>>>(ws + WS_H1 + 0,    PF(P_AM_W2), PF(P_AM_B2), ws + WS_H2 + 0,    4096, 4096, 1);
  gemv_kernel<<<512, 256, 0, stream>>>(ws + WS_H1 + 4096, PF(P_AE_W2), PF(P_AE_B2), ws + WS_H2 + 4096, 4096, 4096, 1);
  gemv_kernel<<<512, 256, 0, stream>>>(ws + WS_H1 + 8192, PF(P_CR_W2), PF(P_CR_B2), ws + WS_H2 + 8192, 4096, 4096, 1);

  // 6. final projections + masked softmax + value
  final_kernel<<<1, 256, 0, stream>>>(P, ws, out);
}